// Pointnet2Backbone_28595892257574
// MI455X (gfx1250) — compile-verified
//
#include <hip/hip_runtime.h>
#include <hip/hip_bf16.h>

typedef float v2f __attribute__((ext_vector_type(2)));
typedef float v8f __attribute__((ext_vector_type(8)));

// ---------------------------------------------------------------------------
// Farthest point sampling: one block per batch. mind[] kept in VGPRs.
// Writes selected coords (x_out) and indices (i_out) directly.
// ---------------------------------------------------------------------------
template <int N, int NPOINT, int T>
__global__ __launch_bounds__(T) void fps_kernel(const float* __restrict__ src,
                                                int stride,
                                                float* __restrict__ x_out,
                                                int* __restrict__ i_out) {
  constexpr int K = N / T;
  constexpr int NW = T / 32;
  __shared__ float red_v[NW];
  __shared__ int red_i[NW];
  __shared__ float sp[3];
  const int b = blockIdx.x;
  const int tid = threadIdx.x;
  const int lane = tid & 31;
  const int wv = tid >> 5;
  const float* base = src + (size_t)b * N * stride;

  float mind[K];
#pragma unroll
  for (int k = 0; k < K; ++k) mind[k] = 1e10f;

  if (tid == 0) {
    i_out[b * NPOINT] = 0;
    float px = base[0], py = base[1], pz = base[2];
    sp[0] = px; sp[1] = py; sp[2] = pz;
    float* xo = x_out + (size_t)b * NPOINT * 3;
    xo[0] = px; xo[1] = py; xo[2] = pz;
  }
  __syncthreads();

  for (int s = 1; s < NPOINT; ++s) {
    const float px = sp[0], py = sp[1], pz = sp[2];
    float best = -1.0f;
    int bi = 0;
#pragma unroll
    for (int k = 0; k < K; ++k) {
      const int i = k * T + tid;
      const float* p = base + (size_t)i * stride;
      const float dx = p[0] - px, dy = p[1] - py, dz = p[2] - pz;
      const float d = dx * dx + dy * dy + dz * dz;
      const float m = fminf(mind[k], d);
      mind[k] = m;
      if (m > best) { best = m; bi = i; }
    }
    // wave32 argmax reduce (first-max tie break: smaller index wins on ties)
#pragma unroll
    for (int off = 16; off > 0; off >>= 1) {
      const float v2 = __shfl_down(best, off, 32);
      const int i2 = __shfl_down(bi, off, 32);
      if (v2 > best || (v2 == best && i2 < bi)) { best = v2; bi = i2; }
    }
    if (lane == 0) { red_v[wv] = best; red_i[wv] = bi; }
    __syncthreads();
    if (tid < 32) {
      float v = (tid < NW) ? red_v[tid] : -2.0f;
      int i = (tid < NW) ? red_i[tid] : 0;
#pragma unroll
      for (int off = 16; off > 0; off >>= 1) {
        const float v2 = __shfl_down(v, off, 32);
        const int i2 = __shfl_down(i, off, 32);
        if (v2 > v || (v2 == v && i2 < i)) { v = v2; i = i2; }
      }
      if (tid == 0) {
        i_out[b * NPOINT + s] = i;
        const float* p = base + (size_t)i * stride;
        const float qx = p[0], qy = p[1], qz = p[2];
        sp[0] = qx; sp[1] = qy; sp[2] = qz;
        float* xo = x_out + ((size_t)b * NPOINT + s) * 3;
        xo[0] = qx; xo[1] = qy; xo[2] = qz;
      }
    }
    __syncthreads();
  }
}

// ---------------------------------------------------------------------------
// Ball query: one wave32 per center. Collect first NS in-radius indices in
// ascending index order via ballot + prefix popcount; pad with first hit.
// ---------------------------------------------------------------------------
template <int N, int NS>
__global__ __launch_bounds__(128) void ballq_kernel(
    const float* __restrict__ src, int stride, const float* __restrict__ centers,
    int S, int* __restrict__ idx_out, float r2) {
  const int lane = threadIdx.x & 31;
  const int gc = blockIdx.x * (blockDim.x >> 5) + (threadIdx.x >> 5);
  const int b = gc / S;
  const float* base = src + (size_t)b * N * stride;
  const float* c = centers + (size_t)gc * 3;
  const float cx = c[0], cy = c[1], cz = c[2];
  int* out = idx_out + (size_t)gc * NS;

  int cnt = 0;
  int first = 0;
  bool have_first = false;
  for (int b0 = 0; b0 < N; b0 += 32) {
    const int i = b0 + lane;
    bool inb = false;
    if (i < N) {
      const float* p = base + (size_t)i * stride;
      const float dx = p[0] - cx, dy = p[1] - cy, dz = p[2] - cz;
      inb = (dx * dx + dy * dy + dz * dz) < r2;
    }
    const unsigned m = __builtin_amdgcn_ballot_w32(inb);
    if (m) {
      if (!have_first) { first = b0 + (__ffs(m) - 1); have_first = true; }
      if (inb) {
        const int pos = cnt + __popc(m & ((1u << lane) - 1u));
        if (pos < NS) out[pos] = i;
      }
      cnt += __popc(m);
      if (cnt >= NS) break;
    }
  }
  for (int j = cnt + lane; j < NS; j += 32) out[j] = first;
}

// ---------------------------------------------------------------------------
// Zero-pad layer-1 weights from (cin x cout) to (cinp x cout) in workspace so
// the WMMA inner loop needs no lane-divergent guards.
// ---------------------------------------------------------------------------
__global__ void pad_w_kernel(const float* __restrict__ W, float* __restrict__ Wp,
                             int cin, int cout, int total) {
  const int i = blockIdx.x * blockDim.x + threadIdx.x;
  if (i < total) {
    const int r = i / cout;
    Wp[i] = (r < cin) ? W[i] : 0.0f;
  }
}

// ---------------------------------------------------------------------------
// One MLP layer (all shapes compile-time, weights pre-padded to CINP rows):
// LDS(in) x global W -> LDS(out) via V_WMMA_F32_16X16X4_F32, K fully unrolled,
// branch-free inner loop.
// A 16x4 f32: lane<16 -> K0/K1, lane>=16 -> K2/K3 of rows 0..15.
// B 4x16 f32: lanes 0-15 -> K0/K1, lanes 16-31 -> K2/K3, N striped on lanes.
// C/D: col = 16*nt + (lane&15), row = 16*mt + v + 8*(lane>=16).
// ---------------------------------------------------------------------------
template <int ROWS, int CINP, int COUT, int LDA, int LDB>
__device__ __forceinline__ void mlp_layer(const float* in,
                                          const float* __restrict__ W,
                                          const float* __restrict__ bias,
                                          float* out, int tid) {
  const int lane = tid & 31;
  const int wave = tid >> 5;
  const int hl = lane >> 4;  // 0 or 1
  const int lm = lane & 15;
  constexpr int MT = ROWS / 16;
  constexpr int NT = COUT / 16;
  for (int t = wave; t < MT * NT; t += 4) {
    const int mt = t / NT, nt = t % NT;
    const float* arow = in + (mt * 16 + lm) * LDA;
    const int col = nt * 16 + lm;
    const float* wcol = W + col + hl * 2 * COUT;
    v8f acc = {};
#pragma unroll
    for (int k0 = 0; k0 < CINP; k0 += 4) {
      const int kk = k0 + hl * 2;
      v2f a, bb;
      a.x = arow[kk];
      a.y = arow[kk + 1];
      bb.x = wcol[k0 * COUT];
      bb.y = wcol[(k0 + 1) * COUT];
      acc = __builtin_amdgcn_wmma_f32_16x16x4_f32(false, a, false, bb, (short)0,
                                                  acc, false, false);
    }
    const float bv = bias[col];
    float* orow = out + mt * 16 * LDB + col;
#pragma unroll
    for (int v = 0; v < 8; ++v) {
      orow[(v + hl * 8) * LDB] = fmaxf(acc[v] + bv, 0.0f);
    }
  }
}

// ---------------------------------------------------------------------------
// Fused gather + 3-layer WMMA MLP + max-pool over nsample.
// ---------------------------------------------------------------------------
template <int NS, int CPB, int CIN, int CINP, int C1, int C2, int C3, int BA,
          int BB>
__global__ __launch_bounds__(128) void mlp_kernel(
    const float* __restrict__ pts, int pstride, const float* __restrict__ feats,
    int fstride, const float* __restrict__ centers, const int* __restrict__ idx,
    const float* __restrict__ W1p, const float* __restrict__ Bb1,
    const float* __restrict__ W2, const float* __restrict__ Bb2,
    const float* __restrict__ W3, const float* __restrict__ Bb3,
    float* __restrict__ f_out, int S, int N, float inv_r) {
  constexpr int ROWS = NS * CPB;
  __shared__ float bufA[ROWS * BA];
  __shared__ float bufB[ROWS * BB];
  const int tid = threadIdx.x;
  const int gc0 = blockIdx.x * CPB;

  // Build h0 = [(gathered xyz - center)/radius, gathered feats, 0-pad]
  for (int e = tid; e < ROWS * CINP; e += 128) {
    const int r = e / CINP, c = e % CINP;
    const int lc = r / NS, j = r % NS;
    const int gc = gc0 + lc;
    const int b = gc / S;
    const int pid = idx[(size_t)gc * NS + j];
    float v = 0.0f;
    if (c < 3) {
      v = (pts[((size_t)b * N + pid) * pstride + c] -
           centers[(size_t)gc * 3 + c]) * inv_r;
    } else if (c < CIN) {
      v = feats[((size_t)b * N + pid) * fstride + (c - 3)];
    }
    bufA[r * BA + c] = v;
  }
  __syncthreads();
  mlp_layer<ROWS, CINP, C1, BA, BB>(bufA, W1p, Bb1, bufB, tid);
  __syncthreads();
  mlp_layer<ROWS, C1, C2, BB, BA>(bufB, W2, Bb2, bufA, tid);
  __syncthreads();
  mlp_layer<ROWS, C2, C3, BA, BB>(bufA, W3, Bb3, bufB, tid);
  __syncthreads();
  // max over nsample
  for (int o = tid; o < CPB * C3; o += 128) {
    const int lc = o / C3, c = o % C3;
    float m = -1e30f;
    for (int j = 0; j < NS; ++j) m = fmaxf(m, bufB[(lc * NS + j) * BB + c]);
    f_out[(size_t)(gc0 + lc) * C3 + c] = m;
  }
}

// ---------------------------------------------------------------------------
extern "C" void kernel_launch(void* const* d_in, const int* in_sizes, int n_in,
                              void* d_out, int out_size, void* d_ws,
                              size_t ws_size, hipStream_t stream) {
  (void)in_sizes; (void)n_in; (void)out_size; (void)ws_size;
  const float* pc = (const float*)d_in[0];
  const float* w10 = (const float*)d_in[1];  const float* b10 = (const float*)d_in[2];
  const float* w11 = (const float*)d_in[3];  const float* b11 = (const float*)d_in[4];
  const float* w12 = (const float*)d_in[5];  const float* b12 = (const float*)d_in[6];
  const float* w20 = (const float*)d_in[7];  const float* b20 = (const float*)d_in[8];
  const float* w21 = (const float*)d_in[9];  const float* b21 = (const float*)d_in[10];
  const float* w22 = (const float*)d_in[11]; const float* b22 = (const float*)d_in[12];
  const float* w30 = (const float*)d_in[13]; const float* b30 = (const float*)d_in[14];
  const float* w31 = (const float*)d_in[15]; const float* b31 = (const float*)d_in[16];
  const float* w32 = (const float*)d_in[17]; const float* b32 = (const float*)d_in[18];

  float* out = (float*)d_out;
  float* x1 = out;                 // 8*512*3   = 12288
  float* f1 = out + 12288;         // 8*512*64  = 262144
  int* i1 = (int*)(out + 274432);  // 8*512     = 4096
  float* x2 = out + 278528;        // 8*128*3   = 3072
  float* f2 = out + 281600;        // 8*128*128 = 131072
  int* i2 = (int*)(out + 412672);  // 8*128     = 1024
  float* x3 = out + 413696;        // 8*64*3    = 1536
  float* f3 = out + 415232;        // 8*64*256  = 131072
  int* i3 = (int*)(out + 546304);  // 8*64      = 512

  int* idx1 = (int*)d_ws;              // 4096*32 ints
  int* idx2 = idx1 + 4096 * 32;        // 1024*32 ints
  int* idx3 = idx2 + 1024 * 32;        // 512*16 ints
  float* w10p = (float*)(idx3 + 512 * 16);  //   8*32  =   256 floats
  float* w20p = w10p + 8 * 32;              //  68*64  =  4352 floats
  float* w30p = w20p + 68 * 64;             // 132*128 = 16896 floats

  // Pad layer-1 weights (independent; head of stream).
  pad_w_kernel<<<1, 256, 0, stream>>>(w10, w10p, 6, 32, 8 * 32);
  pad_w_kernel<<<17, 256, 0, stream>>>(w20, w20p, 67, 64, 68 * 64);
  pad_w_kernel<<<66, 256, 0, stream>>>(w30, w30p, 131, 128, 132 * 128);

  // ---- SA1: N=32768 -> S=512, r=0.2, ns=32, C: 6->32->32->64
  fps_kernel<32768, 512, 1024><<<8, 1024, 0, stream>>>(pc, 6, x1, i1);
  ballq_kernel<32768, 32><<<1024, 128, 0, stream>>>(pc, 6, x1, 512, idx1, 0.04f);
  mlp_kernel<32, 4, 6, 8, 32, 32, 64, 32, 64><<<1024, 128, 0, stream>>>(
      pc, 6, pc + 3, 6, x1, idx1, w10p, b10, w11, b11, w12, b12, f1, 512, 32768,
      5.0f);

  // ---- SA2: N=512 -> S=128, r=0.4, ns=32, C: 67->64->64->128
  fps_kernel<512, 128, 512><<<8, 512, 0, stream>>>(x1, 3, x2, i2);
  ballq_kernel<512, 32><<<256, 128, 0, stream>>>(x1, 3, x2, 128, idx2, 0.16f);
  mlp_kernel<32, 2, 67, 68, 64, 64, 128, 68, 128><<<512, 128, 0, stream>>>(
      x1, 3, f1, 64, x2, idx2, w20p, b20, w21, b21, w22, b22, f2, 128, 512,
      2.5f);

  // ---- SA3: N=128 -> S=64, r=0.8, ns=16, C: 131->128->128->256
  fps_kernel<128, 64, 128><<<8, 128, 0, stream>>>(x2, 3, x3, i3);
  ballq_kernel<128, 16><<<128, 128, 0, stream>>>(x2, 3, x3, 64, idx3, 0.64f);
  mlp_kernel<16, 2, 131, 132, 128, 128, 256, 132, 256><<<256, 128, 0, stream>>>(
      x2, 3, f2, 128, x3, idx3, w30p, b30, w31, b31, w32, b32, f3, 64, 128,
      1.25f);
}